// CustomMeanIoU_73821897883667
// MI455X (gfx1250) — compile-verified
//
#include <hip/hip_runtime.h>
#include <hip/hip_bf16.h>

#define NCLS 19
#define NIMG 16
#define HWPX (512 * 512)          // pixels per (image, class) plane
#define NC_TOT (NIMG * NCLS)      // 304 per-image-per-class counters
#define PIX_PER_THREAD 4
#define BLOCK 256
#define EPSV 1e-6f

typedef __attribute__((ext_vector_type(2))) float v2f;
typedef __attribute__((ext_vector_type(8))) float v8f;

// ---------------------------------------------------------------------------
// Zero the workspace counters (3 * 304 uint32). Deterministic per-launch init.
// ---------------------------------------------------------------------------
__global__ void miou_zero_kernel(unsigned int* __restrict__ cnt, int n) {
    int i = blockIdx.x * blockDim.x + threadIdx.x;
    if (i < n) cnt[i] = 0u;
}

// ---------------------------------------------------------------------------
// Main bandwidth kernel: per-pixel argmax over 19 classes for preds & targets,
// LDS-privatized histograms, one global integer-atomic flush per block.
// Each thread handles 4 consecutive pixels via float4 (b128) loads.
// ---------------------------------------------------------------------------
__global__ __launch_bounds__(BLOCK) void miou_hist_kernel(
    const float* __restrict__ preds,
    const float* __restrict__ targets,
    unsigned int* __restrict__ cnt)   // [NC_TOT pred | NC_TOT targ | NC_TOT inter]
{
    const int n   = blockIdx.y;                                   // image
    const int tid = blockIdx.x * blockDim.x + threadIdx.x;
    const int p   = tid * PIX_PER_THREAD;                         // first pixel

    __shared__ unsigned int h_pred[NCLS];
    __shared__ unsigned int h_targ[NCLS];
    __shared__ unsigned int h_inter[NCLS];
    if (threadIdx.x < NCLS) {
        h_pred[threadIdx.x]  = 0u;
        h_targ[threadIdx.x]  = 0u;
        h_inter[threadIdx.x] = 0u;
    }
    __syncthreads();

    const size_t img_off = (size_t)n * NCLS * HWPX + (size_t)p;
    const float4* pb = (const float4*)(preds   + img_off);
    const float4* tb = (const float4*)(targets + img_off);
    const size_t cstride = HWPX / 4;   // float4 stride between class planes

    // ---- argmax over classes for preds (first-max semantics) ----
    float4 pm = pb[0];
    int pi0 = 0, pi1 = 0, pi2 = 0, pi3 = 0;
#pragma unroll
    for (int c = 1; c < NCLS; ++c) {
        float4 v = pb[(size_t)c * cstride];
        if (v.x > pm.x) { pm.x = v.x; pi0 = c; }
        if (v.y > pm.y) { pm.y = v.y; pi1 = c; }
        if (v.z > pm.z) { pm.z = v.z; pi2 = c; }
        if (v.w > pm.w) { pm.w = v.w; pi3 = c; }
    }

    // ---- argmax over classes for targets ----
    float4 tm = tb[0];
    int ti0 = 0, ti1 = 0, ti2 = 0, ti3 = 0;
#pragma unroll
    for (int c = 1; c < NCLS; ++c) {
        float4 v = tb[(size_t)c * cstride];
        if (v.x > tm.x) { tm.x = v.x; ti0 = c; }
        if (v.y > tm.y) { tm.y = v.y; ti1 = c; }
        if (v.z > tm.z) { tm.z = v.z; ti2 = c; }
        if (v.w > tm.w) { tm.w = v.w; ti3 = c; }
    }

    // ---- LDS histogram (ds_add_u32) ----
    atomicAdd(&h_pred[pi0], 1u); atomicAdd(&h_pred[pi1], 1u);
    atomicAdd(&h_pred[pi2], 1u); atomicAdd(&h_pred[pi3], 1u);
    atomicAdd(&h_targ[ti0], 1u); atomicAdd(&h_targ[ti1], 1u);
    atomicAdd(&h_targ[ti2], 1u); atomicAdd(&h_targ[ti3], 1u);
    if (pi0 == ti0) atomicAdd(&h_inter[pi0], 1u);
    if (pi1 == ti1) atomicAdd(&h_inter[pi1], 1u);
    if (pi2 == ti2) atomicAdd(&h_inter[pi2], 1u);
    if (pi3 == ti3) atomicAdd(&h_inter[pi3], 1u);
    __syncthreads();

    // ---- flush block-private histogram to global per-image counters ----
    if (threadIdx.x < NCLS) {
        int idx = n * NCLS + threadIdx.x;
        atomicAdd(&cnt[idx],              h_pred[threadIdx.x]);
        atomicAdd(&cnt[NC_TOT + idx],     h_targ[threadIdx.x]);
        atomicAdd(&cnt[2 * NC_TOT + idx], h_inter[threadIdx.x]);
    }
}

// ---------------------------------------------------------------------------
// Finalize: one wave32. Compute 304 weighted IoUs, reduce them with
// V_WMMA_F32_16X16X4_F32 against an all-ones B matrix (D = A*1 + C).
// Grand total = sum over M of any column of D.
// ---------------------------------------------------------------------------
__global__ __launch_bounds__(32) void miou_finalize_kernel(
    const unsigned int* __restrict__ cnt,
    const float* __restrict__ class_weights,
    float* __restrict__ out)
{
    __shared__ float vals[320];   // 5 WMMA iterations * 64, zero-padded
    const int lane = threadIdx.x;

    for (int i = lane; i < 320; i += 32) vals[i] = 0.0f;
    __syncthreads();

    for (int i = lane; i < NC_TOT; i += 32) {
        float inter = (float)cnt[2 * NC_TOT + i];
        float uni   = (float)cnt[i] + (float)cnt[NC_TOT + i] - inter;
        float iou   = (inter + EPSV) / (uni + EPSV);
        vals[i] = iou * class_weights[i % NCLS];
    }
    __syncthreads();

    // A fragment layout (32-bit A 16x4): lanes 0-15 hold (M=lane, K=0..1),
    // lanes 16-31 hold (M=lane-16, K=2..3). A[m][k] = vals[it*64 + m*4 + k].
    const int m     = (lane < 16) ? lane : (lane - 16);
    const int kbase = (lane < 16) ? 0 : 2;

    v2f b; b.x = 1.0f; b.y = 1.0f;   // all-ones B: layout-independent
    v8f acc = {};                    // zero accumulator

#pragma unroll
    for (int it = 0; it < 5; ++it) {
        const int off = it * 64 + m * 4 + kbase;
        v2f a; a.x = vals[off]; a.y = vals[off + 1];
        acc = __builtin_amdgcn_wmma_f32_16x16x4_f32(
            /*neg_a=*/false, a, /*neg_b=*/false, b,
            /*c_mod=*/(short)0, acc, /*reuse_a=*/false, /*reuse_b=*/false);
    }

    // D layout: VGPR r, lane 0 -> D[M=r][N=0]; lane 16 -> D[M=8+r][N=0].
    float total = 0.0f;
#pragma unroll
    for (int r = 0; r < 8; ++r) {
        total += __shfl(acc[r], 0, 32);
        total += __shfl(acc[r], 16, 32);
    }

    if (lane == 0) out[0] = total / (float)NC_TOT;   // mean over C then N
}

// ---------------------------------------------------------------------------
extern "C" void kernel_launch(void* const* d_in, const int* in_sizes, int n_in,
                              void* d_out, int out_size, void* d_ws, size_t ws_size,
                              hipStream_t stream) {
    const float* preds   = (const float*)d_in[0];
    const float* targets = (const float*)d_in[1];
    const float* cw      = (const float*)d_in[2];
    float* out           = (float*)d_out;
    unsigned int* cnt    = (unsigned int*)d_ws;   // 3 * NC_TOT uint32

    const int n_cnt = 3 * NC_TOT;
    miou_zero_kernel<<<(n_cnt + 255) / 256, 256, 0, stream>>>(cnt, n_cnt);

    dim3 grid(HWPX / (BLOCK * PIX_PER_THREAD), NIMG);   // (256, 16)
    miou_hist_kernel<<<grid, BLOCK, 0, stream>>>(preds, targets, cnt);

    miou_finalize_kernel<<<1, 32, 0, stream>>>(cnt, cw, out);
}